// SpikingNeuralNet_82188494176428
// MI455X (gfx1250) — compile-verified
//
#include <hip/hip_runtime.h>
#include <stdint.h>

// ---------------------------------------------------------------------------
// SpikingNeuralNet on MI455X (gfx1250): single persistent WGP workgroup,
// wave32 shuffle argmax, async global->LDS row fetch overlapped with Threefry.
// ---------------------------------------------------------------------------

#define ALPHA_C   3.0f
#define V_RESET_C 1.0f
#define NMAX      4096
#define BLK       1024
#define NPT       4          // NMAX / BLK neurons per thread

// ---- gfx1250 async global->LDS path (guarded; fallback = plain loads) -----
#if defined(__has_builtin)
#if __has_builtin(__builtin_amdgcn_global_load_async_to_lds_b128) && \
    __has_builtin(__builtin_amdgcn_s_wait_asynccnt)
#define HAVE_ASYNC_LDS 1
#endif
#endif

#ifdef HAVE_ASYNC_LDS
typedef int v4i __attribute__((ext_vector_type(4)));
typedef __attribute__((address_space(1))) v4i* g_v4i_p;   // global (as1)
typedef __attribute__((address_space(3))) v4i* l_v4i_p;   // LDS (as3)
#endif

// ---- Threefry-2x32 (bit-exact with jax.random) ----------------------------
struct U2 { uint32_t x, y; };

__device__ __forceinline__ uint32_t rotl32(uint32_t v, int r) {
  return (v << r) | (v >> (32 - r));
}

__device__ __forceinline__ U2 threefry2x32(uint32_t k0, uint32_t k1,
                                           uint32_t x0, uint32_t x1) {
  uint32_t ks2 = k0 ^ k1 ^ 0x1BD11BDAu;
  x0 += k0; x1 += k1;
#define TF_ROUND(r) { x0 += x1; x1 = rotl32(x1, (r)); x1 ^= x0; }
  TF_ROUND(13) TF_ROUND(15) TF_ROUND(26) TF_ROUND(6)
  x0 += k1;  x1 += ks2 + 1u;
  TF_ROUND(17) TF_ROUND(29) TF_ROUND(16) TF_ROUND(24)
  x0 += ks2; x1 += k0 + 2u;
  TF_ROUND(13) TF_ROUND(15) TF_ROUND(26) TF_ROUND(6)
  x0 += k0;  x1 += k1 + 3u;
  TF_ROUND(17) TF_ROUND(29) TF_ROUND(16) TF_ROUND(24)
  x0 += k1;  x1 += ks2 + 4u;
  TF_ROUND(13) TF_ROUND(15) TF_ROUND(26) TF_ROUND(6)
  x0 += ks2; x1 += k0 + 5u;
#undef TF_ROUND
  U2 r; r.x = x0; r.y = x1; return r;
}

__device__ __forceinline__ float bits_to_uniform(uint32_t b) {
  // jax: bitcast((bits >> 9) | 0x3f800000) - 1.0
  return __uint_as_float((b >> 9) | 0x3f800000u) - 1.0f;
}

__device__ __forceinline__ float softplus_f(float x) {
  // jax.nn.softplus = max(x,0) + log1p(exp(-|x|))
  return fmaxf(x, 0.0f) + log1pf(expf(-fabsf(x)));
}

// ---- init kernel: tev/yev must start at +inf (harness poisons d_out) ------
__global__ void snn_init_kernel(float* __restrict__ out, const int* __restrict__ p_ms,
                                int N, int T) {
  const int ms = *p_ms;
  const size_t base  = (size_t)T * (size_t)N * 3;               // skip ys
  const size_t total = (size_t)ms + (size_t)ms * (size_t)N * 3; // tev + yev
  const size_t stride = (size_t)gridDim.x * blockDim.x;
  const float inf = __int_as_float(0x7f800000);
  for (size_t i = (size_t)blockIdx.x * blockDim.x + threadIdx.x; i < total; i += stride)
    out[base + i] = inf;
}

// ---- main persistent kernel ----------------------------------------------
__global__ __launch_bounds__(BLK, 1)
void snn_main_kernel(const float* __restrict__ w,
                     const float* __restrict__ mu,
                     const float* __restrict__ I_ext,
                     const float* __restrict__ ts,
                     const float* __restrict__ v0,
                     const float* __restrict__ i0,
                     const int*   __restrict__ p_ms,
                     const int*   __restrict__ p_seed,
                     float* __restrict__ out,
                     int N, int T) {
  const int tid = threadIdx.x;
  const int ms = *p_ms;
  const uint32_t seed = (uint32_t)(*p_seed);

  float* out_ys  = out;
  float* out_tev = out + (size_t)T * (size_t)N * 3;
  float* out_yev = out_tev + (size_t)ms;
  int*   out_nsp = (int*)(out_yev + (size_t)ms * (size_t)N * 3);

  __shared__ __align__(16) float s_wrow[NMAX];
  __shared__ float s_redv[32];
  __shared__ int   s_redi[32];
  __shared__ float s_bcv;     // max of s_new
  __shared__ int   s_bci;     // argmax index
  __shared__ float s_sold;    // s_old[idx]

  // key = jax.random.key(seed) -> [0, seed]; split -> (init_key, trans_key)
  U2 a = threefry2x32(0u, seed, 0u, 2u);
  U2 b = threefry2x32(0u, seed, 1u, 3u);
  const uint32_t ik0 = a.x, ik1 = b.x;   // init_key
  const uint32_t tk0 = a.y, tk1 = b.y;   // trans_key

  const float mu0 = mu[0], mu1 = mu[1];
  const float dt  = ts[1] - ts[0];
  const int   half = N >> 1;
  const int   nbase = tid * NPT;

  float v[NPT], cu[NPT], s[NPT], Ie[NPT];
#pragma unroll
  for (int j = 0; j < NPT; ++j) {
    const int n = nbase + j;
    const bool ok = n < N;
    v[j]  = ok ? v0[n]    : 0.0f;
    cu[j] = ok ? i0[n]    : 0.0f;
    Ie[j] = ok ? I_ext[n] : 0.0f;
    if (ok) {
      U2 kn = threefry2x32(ik0, ik1, 0u, (uint32_t)n);       // fold_in(init_key, n)
      uint32_t bits = threefry2x32(kn.x, kn.y, 0u, 0u).x;    // uniform(key, ())
      s[j] = logf(bits_to_uniform(bits)) - ALPHA_C;
    } else {
      s[j] = -__builtin_inff();
    }
  }

  // ys[0] = y0
#pragma unroll
  for (int j = 0; j < NPT; ++j) {
    const int n = nbase + j;
    if (n < N) {
      float* p = out_ys + (size_t)n * 3;
      __builtin_nontemporal_store(v[j],  p + 0);
      __builtin_nontemporal_store(cu[j], p + 1);
      __builtin_nontemporal_store(s[j],  p + 2);
    }
  }

  int nsp = 0;
  for (int k = 0; k < T - 1; ++k) {
    const float t = ts[k];

    // Euler step
    float vn[NPT], in_[NPT], sn[NPT];
#pragma unroll
    for (int j = 0; j < NPT; ++j) {
      const float dv = mu0 * (cu[j] - v[j]) + mu0 * Ie[j];
      const float di = -mu1 * cu[j];
      const float ds = softplus_f(v[j]);
      vn[j]  = v[j]  + dt * dv;
      in_[j] = cu[j] + dt * di;
      sn[j]  = s[j]  + dt * ds;
    }

    // local argmax of s_new (ties -> smaller index, like jnp.argmax)
    float lmax = sn[0]; int lidx = nbase;
#pragma unroll
    for (int j = 1; j < NPT; ++j)
      if (sn[j] > lmax) { lmax = sn[j]; lidx = nbase + j; }

    // wave32 shuffle reduction
#pragma unroll
    for (int off = 16; off >= 1; off >>= 1) {
      float ov = __shfl_xor(lmax, off, 32);
      int   oi = __shfl_xor(lidx, off, 32);
      if (ov > lmax || (ov == lmax && oi < lidx)) { lmax = ov; lidx = oi; }
    }
    if ((tid & 31) == 0) { s_redv[tid >> 5] = lmax; s_redi[tid >> 5] = lidx; }
    __syncthreads();
    if (tid < 32) {
      float rv = s_redv[tid]; int ri = s_redi[tid];
#pragma unroll
      for (int off = 16; off >= 1; off >>= 1) {
        float ov = __shfl_xor(rv, off, 32);
        int   oi = __shfl_xor(ri, off, 32);
        if (ov > rv || (ov == rv && oi < ri)) { rv = ov; ri = oi; }
      }
      if (tid == 0) { s_bcv = rv; s_bci = ri; }
    }
    __syncthreads();
    const float smax = s_bcv;
    const int   idx  = s_bci;
    const bool  spike = (smax >= 0.0f);     // uniform across the workgroup

    if (spike) {
      // owner publishes s_old[idx] (s_new[idx] == smax already broadcast)
      if (idx >= nbase && idx < nbase + NPT) s_sold = s[idx - nbase];

      // kick off async copy of our 16B slice of w[idx,:] into LDS
#ifdef HAVE_ASYNC_LDS
      {
        const float* gsrc = w + (size_t)idx * (size_t)N + nbase;
        g_v4i_p g = (g_v4i_p)(uintptr_t)gsrc;
        l_v4i_p l = (l_v4i_p)(uint32_t)(uintptr_t)&s_wrow[nbase];
        __builtin_amdgcn_global_load_async_to_lds_b128(g, l, 0, 0);
      }
#endif
      // overlap: Threefry draws for this step (fold_in(trans_key, k) chain)
      U2 sk = threefry2x32(tk0, tk1, 0u, (uint32_t)k);
      float u[NPT];
#pragma unroll
      for (int j = 0; j < NPT; ++j) {
        const int n = nbase + j;
        uint32_t bits;
        if (n < half) bits = threefry2x32(sk.x, sk.y, (uint32_t)n, (uint32_t)(n + half)).x;
        else          bits = threefry2x32(sk.x, sk.y, (uint32_t)(n - half), (uint32_t)n).y;
        u[j] = bits_to_uniform(bits);
      }

      __syncthreads();                       // s_sold visible
      const float sold  = s_sold;
      const float denom = sold - smax;
      float frac = (denom == 0.0f) ? 1.0f : fminf(fmaxf(sold / denom, 0.0f), 1.0f);
      const float tevent = t + frac * dt;

#ifdef HAVE_ASYNC_LDS
      __builtin_amdgcn_s_wait_asynccnt(0);   // our slice of w-row is in LDS
      __asm__ __volatile__("" ::: "memory");
#endif

      float yev_[NPT][3];
#pragma unroll
      for (int j = 0; j < NPT; ++j) {
        const float yv  = v[j]  + frac * (vn[j]  - v[j]);
        const float yi  = cu[j] + frac * (in_[j] - cu[j]);
        const float ysv = s[j]  + frac * (sn[j]  - s[j]);
        yev_[j][0] = yv; yev_[j][1] = yi; yev_[j][2] = ysv;
        const bool ev = (sn[j] >= 0.0f);
        float wv;
#ifdef HAVE_ASYNC_LDS
        wv = s_wrow[nbase + j];
#else
        wv = w[(size_t)idx * (size_t)N + nbase + j];
#endif
        v[j]  = ev ? (yv - V_RESET_C) : yv;
        cu[j] = ev ? yi : (yi + wv);
        s[j]  = ev ? (logf(u[j]) - ALPHA_C) : ysv;
      }

      if (nsp < ms) {
        if (tid == 0) out_tev[nsp] = tevent;
        float* dst = out_yev + ((size_t)nsp * (size_t)N + nbase) * 3;
#pragma unroll
        for (int j = 0; j < NPT; ++j) {
          if (nbase + j < N) {
            __builtin_nontemporal_store(yev_[j][0], dst + j * 3 + 0);
            __builtin_nontemporal_store(yev_[j][1], dst + j * 3 + 1);
            __builtin_nontemporal_store(yev_[j][2], dst + j * 3 + 2);
          }
        }
      }
      ++nsp;
    } else {
#pragma unroll
      for (int j = 0; j < NPT; ++j) { v[j] = vn[j]; cu[j] = in_[j]; s[j] = sn[j]; }
    }

    // ys[k+1] = y_out  (streaming, non-temporal: keep w resident in L2)
    float* row = out_ys + ((size_t)(k + 1) * (size_t)N + nbase) * 3;
#pragma unroll
    for (int j = 0; j < NPT; ++j) {
      if (nbase + j < N) {
        __builtin_nontemporal_store(v[j],  row + j * 3 + 0);
        __builtin_nontemporal_store(cu[j], row + j * 3 + 1);
        __builtin_nontemporal_store(s[j],  row + j * 3 + 2);
      }
    }
  }

  if (tid == 0) *out_nsp = nsp;
}

extern "C" void kernel_launch(void* const* d_in, const int* in_sizes, int n_in,
                              void* d_out, int out_size, void* d_ws, size_t ws_size,
                              hipStream_t stream) {
  (void)n_in; (void)out_size; (void)d_ws; (void)ws_size;
  const float* w     = (const float*)d_in[0];
  const float* mu    = (const float*)d_in[1];
  const float* I_ext = (const float*)d_in[2];
  const float* ts    = (const float*)d_in[3];
  const float* v0    = (const float*)d_in[4];
  const float* i0    = (const float*)d_in[5];
  const int*   ms    = (const int*)d_in[6];
  const int*   seed  = (const int*)d_in[7];
  const int N = in_sizes[2];   // I_ext length
  const int T = in_sizes[3];   // ts length

  snn_init_kernel<<<2048, 256, 0, stream>>>((float*)d_out, ms, N, T);
  snn_main_kernel<<<1, BLK, 0, stream>>>(w, mu, I_ext, ts, v0, i0, ms, seed,
                                         (float*)d_out, N, T);
}